// CTCLoss_85590108274767
// MI455X (gfx1250) — compile-verified
//
#include <hip/hip_runtime.h>

#define Tdim 1024
#define Ndim 64
#define Cdim 128
#define Sdim 128
#define SX   257      // 2*S+1 extended states
#define K    9        // states per lane: 32*9 = 288 >= 257

#define NEG_INF (-__builtin_inff())

// logsumexp of 3 values; x is always finite here (alpha is never -inf thanks to
// the reference's zeros-init), y/z may be -inf (exp(-inf)=0 on HW).
__device__ __forceinline__ float lse3(float x, float y, float z) {
    float m = fmaxf(x, fmaxf(y, z));
    return m + __logf(__expf(x - m) + __expf(y - m) + __expf(z - m));
}

// CDNA5 async global->LDS copy (ASYNCcnt-tracked). Each lane moves 16B.
// VDST operand = LDS byte offset (low 32 bits of generic shared pointer).
__device__ __forceinline__ void async_slice_load(const float* g, const float* lds) {
    unsigned ldsoff = (unsigned)(unsigned long long)lds;
    unsigned long long ga = (unsigned long long)g;
    asm volatile("global_load_async_to_lds_b128 %0, %1, off"
                 :: "v"(ldsoff), "v"(ga) : "memory");
}
#define WAIT_ASYNC0() asm volatile("s_wait_asynccnt 0x0" ::: "memory")
#define WAIT_ASYNC1() asm volatile("s_wait_asynccnt 0x1" ::: "memory")

// One wave32 per batch row n. Alpha lives in registers (9 states/lane),
// neighbor states cross lanes via shfl_up, emission slice double-buffered in
// LDS via async tensor-path loads. Barrier-free.
__global__ __launch_bounds__(32) void ctc_alpha_kernel(
    const float* __restrict__ lp,        // [T, N, C] log-probs
    const long long* __restrict__ tgt,   // [N, S] int64 labels (1..C-1)
    const int* __restrict__ ilen,        // [N]
    const int* __restrict__ tlen,        // [N]
    float* __restrict__ v)               // [N] out: alpha[il-1, 2*tl-1]
{
    __shared__ float sl[2][Cdim];        // 2 x 512B emission slices
    const int n     = blockIdx.x;
    const int lane  = threadIdx.x;
    const int Tn    = ilen[n];           // >= 512 per setup
    const int sStar = 2 * tlen[n] - 1;   // in [127, 255]
    const int base  = lane * K;

    // Per-slot class index (ext labels) and skip mask, time-invariant.
    int  cls[K];
    bool skp[K];
    #pragma unroll
    for (int k = 0; k < K; ++k) {
        int s = base + k;
        int c = 0; bool sk = false;
        if (s < SX && (s & 1)) {                 // odd state: real label
            int j = (s - 1) >> 1;
            c = (int)tgt[n * Sdim + j];
            if (s >= 3) sk = (c != (int)tgt[n * Sdim + j - 1]);
        }
        cls[k] = c; skp[k] = sk;                 // even states: blank, no skip
    }

    const float* gbase   = lp + (size_t)n * Cdim + (size_t)lane * 4;
    const size_t tstride = (size_t)Ndim * Cdim;

    // Prime the double buffer: slice 0 (+ prefetch slice 1).
    async_slice_load(gbase, &sl[0][lane * 4]);
    if (Tn > 1) {
        async_slice_load(gbase + tstride, &sl[1][lane * 4]);
        WAIT_ASYNC1();                            // slice 0 landed
    } else {
        WAIT_ASYNC0();
    }

    // Faithful init: alpha0 = 0 everywhere, states 0,1 = emit[0].
    float a[K];
    #pragma unroll
    for (int k = 0; k < K; ++k) {
        int s = base + k;
        float e = sl[0][cls[k]];
        a[k] = (s < 2) ? e : 0.0f;
    }
    if (Tn == 1 && sStar >= base && sStar < base + K) v[n] = a[sStar - base];

    for (int t = 1; t < Tn; ++t) {
        // Prefetch t+1 into the other buffer, then wait for slice t.
        if (t + 1 < Tn) {
            async_slice_load(gbase + (size_t)(t + 1) * tstride,
                             &sl[(t + 1) & 1][lane * 4]);
            WAIT_ASYNC1();
        } else {
            WAIT_ASYNC0();
        }
        const float* e = sl[t & 1];

        float pl  = __shfl_up(a[K - 1], 1);      // alpha[base-1]
        float pl2 = __shfl_up(a[K - 2], 1);      // alpha[base-2]
        if (lane == 0) { pl = NEG_INF; pl2 = NEG_INF; }

        // Descending k: new a[k] reads old a[k-1], a[k-2] (not yet clobbered).
        #pragma unroll
        for (int k = K - 1; k >= 2; --k)
            a[k] = lse3(a[k], a[k - 1], skp[k] ? a[k - 2] : NEG_INF) + e[cls[k]];
        a[1] = lse3(a[1], a[0], skp[1] ? pl  : NEG_INF) + e[cls[1]];
        a[0] = lse3(a[0], pl,   skp[0] ? pl2 : NEG_INF) + e[cls[0]];

        if (t == Tn - 1 && sStar >= base && sStar < base + K)
            v[n] = a[sStar - base];              // exactly one lane writes
    }
}

// Final scalar: -logsumexp over the 64 per-row values. One wave32.
__global__ __launch_bounds__(32) void ctc_reduce_kernel(const float* __restrict__ v,
                                                        float* __restrict__ out) {
    int lane = threadIdx.x;
    float x0 = v[lane], x1 = v[lane + 32];
    float m = fmaxf(x0, x1);
    #pragma unroll
    for (int off = 16; off; off >>= 1) m = fmaxf(m, __shfl_xor(m, off));
    float s = __expf(x0 - m) + __expf(x1 - m);
    #pragma unroll
    for (int off = 16; off; off >>= 1) s += __shfl_xor(s, off);
    if (lane == 0) out[0] = -(m + __logf(s));
}

extern "C" void kernel_launch(void* const* d_in, const int* in_sizes, int n_in,
                              void* d_out, int out_size, void* d_ws, size_t ws_size,
                              hipStream_t stream) {
    const float*     lp   = (const float*)d_in[0];      // [1024,64,128] f32
    const long long* tgt  = (const long long*)d_in[1];  // [64,128] int64
    const int*       ilen = (const int*)d_in[2];        // [64] int32
    const int*       tlen = (const int*)d_in[3];        // [64] int32
    float*           v    = (float*)d_ws;               // [64] scratch
    float*           out  = (float*)d_out;              // scalar f32

    ctc_alpha_kernel<<<dim3(Ndim), dim3(32), 0, stream>>>(lp, tgt, ilen, tlen, v);
    ctc_reduce_kernel<<<dim3(1), dim3(32), 0, stream>>>(v, out);
}